// DecoderRNN_27779848471231
// MI455X (gfx1250) — compile-verified
//
#include <hip/hip_runtime.h>
#include <hip/hip_bf16.h>

// ---------------------------------------------------------------------------
// Show-attend-tell decoder on MI455X (gfx1250, wave32).
// - All GEMMs: V_WMMA_F32_16X16X4_F32 (fp32 in/out). Per-step GEMMs are M=32
//   weight streams resident in the 192MB L2, so fp32 WMMA costs nothing vs
//   f16 here and keeps the 20-step recurrence at reference precision.
// - The one-time enc_proj GEMM (6272x512, K=2048) uses async-to-LDS double
//   buffering (GLOBAL_LOAD_ASYNC_TO_LDS_B128 + s_wait_asynccnt) to share the
//   A-strip across 8 waves and overlap DMA with WMMA.
// ---------------------------------------------------------------------------

typedef float v2f __attribute__((ext_vector_type(2)));
typedef float v8f __attribute__((ext_vector_type(8)));
typedef int   v4i __attribute__((ext_vector_type(4)));

#define B_   32
#define P_   196
#define ENC_ 2048
#define E_   512
#define H_   512
#define A_   512
#define V_   10000
#define T_   21
#define TS_  20   // T-1 decode steps

#if __has_builtin(__builtin_amdgcn_global_load_async_to_lds_b128) && \
    __has_builtin(__builtin_amdgcn_s_wait_asynccnt)
#define ASYNC_LDS 1
#endif

__device__ __forceinline__ v8f wmma4(v2f a, v2f b, v8f c) {
  // 8 args: (neg_a, A, neg_b, B, c_mod, C, reuse_a, reuse_b)
  return __builtin_amdgcn_wmma_f32_16x16x4_f32(
      false, a, false, b, (short)0, c, false, false);
}

__device__ __forceinline__ float sigmoidf_(float x) {
  return 1.0f / (1.0f + __expf(-x));
}

// ---------------------------------------------------------------------------
// Generic GEMM: C[m,n] = sum_k A[m,k] * W[n,k] + bias[n]
// A: MxK row-major, W: NxK row-major, C: row stride ldc.
// One wave per 16x16 output tile; K stepped by 4 per WMMA.
// Fragment layouts per CDNA5 ISA 7.12.2:
//   A 16x4 : lane L holds M=L%16, K = 2*(L/16) + {0,1} in 2 VGPRs
//   B 4x16 : lane L holds N=L%16, K = 2*(L/16) + {0,1} (mirror of A)
//   C/D    : VGPR v, lane L -> M = v + 8*(L/16), N = L%16
// Requires M%16==0, N%16==0, K%64==0.
// ---------------------------------------------------------------------------
__global__ void gemm_wmma_f32(const float* __restrict__ A,
                              const float* __restrict__ W,
                              const float* __restrict__ bias,
                              float* __restrict__ C,
                              int M, int N, int K, int ldc) {
  const int lane  = threadIdx.x & 31;
  const int wid   = threadIdx.x >> 5;
  const int waves = blockDim.x >> 5;
  const int tile  = blockIdx.x * waves + wid;
  const int tiles_n = N >> 4;
  const int tiles_m = M >> 4;
  if (tile >= tiles_m * tiles_n) return;

  const int tm = tile / tiles_n;
  const int tn = tile % tiles_n;
  const int row   = (tm << 4) + (lane & 15);  // A row
  const int col   = (tn << 4) + (lane & 15);  // W row == output col
  const int khalf = (lane >> 4) << 1;         // 0 or 2

  const float* __restrict__ ap = A + (size_t)row * K + khalf;
  const float* __restrict__ wp = W + (size_t)col * K + khalf;

  v8f acc;
  const float bv = bias[col];
#pragma unroll
  for (int i = 0; i < 8; ++i) acc[i] = bv;

  for (int k0 = 0; k0 < K; k0 += 64) {
    __builtin_prefetch(ap + k0 + 64, 0, 0);
    __builtin_prefetch(wp + k0 + 64, 0, 0);
#pragma unroll
    for (int k = k0; k < k0 + 64; k += 4) {
      v2f a = { ap[k], ap[k + 1] };
      v2f b = { wp[k], wp[k + 1] };
      acc = wmma4(a, b, acc);
    }
  }

  const int rbase = (tm << 4) + ((lane >> 4) << 3);
#pragma unroll
  for (int v = 0; v < 8; ++v)
    C[(size_t)(rbase + v) * ldc + col] = acc[v];
}

// ---------------------------------------------------------------------------
// LDS-staged GEMM for the big one-time enc_proj (large M, tiles_n % 8 == 0).
// Block = 8 waves sharing one 16-row A strip; per 64-K block the strip
// (16x64 f32 = 4KB) is staged into LDS by all 256 threads (one b128 each),
// double buffered and overlapped with WMMA via ASYNCcnt.
// ---------------------------------------------------------------------------
__global__ void gemm_wmma_f32_lds(const float* __restrict__ A,
                                  const float* __restrict__ W,
                                  const float* __restrict__ bias,
                                  float* __restrict__ C,
                                  int M, int N, int K, int ldc) {
  (void)M;
  __shared__ float abuf[2][16 * 64];

  const int lane = threadIdx.x & 31;
  const int wid  = threadIdx.x >> 5;
  const int tiles_n = N >> 4;
  const int nb = tiles_n >> 3;                 // N-blocks of 8 tiles
  const int tm = blockIdx.x / nb;
  const int tn = (blockIdx.x % nb) * 8 + wid;

  const int col   = (tn << 4) + (lane & 15);   // output col / W row
  const int khalf = (lane >> 4) << 1;
  const float* __restrict__ wp = W + (size_t)col * K + khalf;

  // cooperative stage: thread i -> row i/16, float4 column (i%16)*4
  const int r  = threadIdx.x >> 4;
  const int cq = (threadIdx.x & 15) << 2;
  const float* __restrict__ gsrc = A + (size_t)((tm << 4) + r) * K + cq;

#ifdef ASYNC_LDS
#define STAGE(bufidx, koff)                                                   \
  __builtin_amdgcn_global_load_async_to_lds_b128(                             \
      (__attribute__((address_space(1))) v4i*)(gsrc + (koff)),                \
      (__attribute__((address_space(3))) v4i*)(&abuf[bufidx][r * 64 + cq]),   \
      0, 0)
#else
#define STAGE(bufidx, koff)                                                   \
  do { *(float4*)(&abuf[bufidx][r * 64 + cq]) =                               \
           *(const float4*)(gsrc + (koff)); } while (0)
#endif

  v8f acc;
  const float bv = bias[col];
#pragma unroll
  for (int i = 0; i < 8; ++i) acc[i] = bv;

  STAGE(0, 0);  // preload first K-block

  for (int k0 = 0; k0 < K; k0 += 64) {
    const int cur = (k0 >> 6) & 1;
    const bool has_next = (k0 + 64) < K;
    if (has_next) STAGE(cur ^ 1, k0 + 64);     // overlap DMA of next block
#ifdef ASYNC_LDS
    if (has_next) __builtin_amdgcn_s_wait_asynccnt(1);
    else          __builtin_amdgcn_s_wait_asynccnt(0);
#endif
    __syncthreads();                            // buffer `cur` fully landed

    const float* lp = &abuf[cur][(lane & 15) * 64 + khalf];
    __builtin_prefetch(wp + k0 + 64, 0, 0);
#pragma unroll
    for (int k = 0; k < 64; k += 4) {
      v2f a = { lp[k], lp[k + 1] };             // ds_load_b64
      v2f b = { wp[k0 + k], wp[k0 + k + 1] };   // global_load_b64 (L2 stream)
      acc = wmma4(a, b, acc);
    }
    __syncthreads();  // all reads of `cur` done before next overwrite
  }
#undef STAGE

  const int rbase = (tm << 4) + ((lane >> 4) << 3);
#pragma unroll
  for (int v = 0; v < 8; ++v)
    C[(size_t)(rbase + v) * ldc + col] = acc[v];
}

// ---------------------------------------------------------------------------
// Fused LSTM-gates GEMM:
// gates[b,n] = x_t @ W_ih[:, :E].T + context @ W_ih[:, E:].T + h @ W_hh.T
//            + b_ih + b_hh      (M=32, N=2048; one wave per 16x16 tile)
// ---------------------------------------------------------------------------
__global__ void gates_wmma_f32(const float* __restrict__ emb,
                               const int*   __restrict__ captions,
                               const float* __restrict__ context,
                               const float* __restrict__ h,
                               const float* __restrict__ W_ih,
                               const float* __restrict__ W_hh,
                               const float* __restrict__ b_ih,
                               const float* __restrict__ b_hh,
                               float* __restrict__ gates,
                               int t) {
  const int lane  = threadIdx.x & 31;
  const int wid   = threadIdx.x >> 5;
  const int tile  = blockIdx.x * (blockDim.x >> 5) + wid;
  const int tiles_n = (4 * H_) >> 4;           // 128
  if (tile >= 2 * tiles_n) return;

  const int tm = tile / tiles_n;
  const int tn = tile % tiles_n;
  const int row   = (tm << 4) + (lane & 15);   // batch index b
  const int col   = (tn << 4) + (lane & 15);   // gate unit n
  const int khalf = (lane >> 4) << 1;

  const int cap = captions[row * T_ + t];
  const float* __restrict__ xp = emb + (size_t)cap * E_ + khalf;
  const float* __restrict__ cp = context + (size_t)row * ENC_ + khalf;
  const float* __restrict__ hp = h + (size_t)row * H_ + khalf;
  const float* __restrict__ w1 = W_ih + (size_t)col * (E_ + ENC_) + khalf;
  const float* __restrict__ w2 = W_hh + (size_t)col * H_ + khalf;

  v8f acc;
  const float bv = b_ih[col] + b_hh[col];
#pragma unroll
  for (int i = 0; i < 8; ++i) acc[i] = bv;

  // segment 1: x_t (K = 512)
  for (int k0 = 0; k0 < E_; k0 += 64) {
    __builtin_prefetch(w1 + k0 + 64, 0, 0);
#pragma unroll
    for (int k = k0; k < k0 + 64; k += 4) {
      v2f a = { xp[k], xp[k + 1] };
      v2f b = { w1[k], w1[k + 1] };
      acc = wmma4(a, b, acc);
    }
  }
  // segment 2: context (K = 2048), weights continue at column E_
  const float* __restrict__ w1c = w1 + E_;
  for (int k0 = 0; k0 < ENC_; k0 += 64) {
    __builtin_prefetch(w1c + k0 + 64, 0, 0);
    __builtin_prefetch(cp  + k0 + 64, 0, 0);
#pragma unroll
    for (int k = k0; k < k0 + 64; k += 4) {
      v2f a = { cp[k], cp[k + 1] };
      v2f b = { w1c[k], w1c[k + 1] };
      acc = wmma4(a, b, acc);
    }
  }
  // segment 3: recurrent h (K = 512)
  for (int k0 = 0; k0 < H_; k0 += 64) {
    __builtin_prefetch(w2 + k0 + 64, 0, 0);
#pragma unroll
    for (int k = k0; k < k0 + 64; k += 4) {
      v2f a = { hp[k], hp[k + 1] };
      v2f b = { w2[k], w2[k + 1] };
      acc = wmma4(a, b, acc);
    }
  }

  const int rbase = (tm << 4) + ((lane >> 4) << 3);
#pragma unroll
  for (int v = 0; v < 8; ++v)
    gates[(size_t)(rbase + v) * (4 * H_) + col] = acc[v];
}

// ---------------------------------------------------------------------------
// att[b,p] = sum_a tanh(enc_proj[b,p,a] + dec_proj[b,a]) * att_W[a] + att_b
// One wave per (b,p); lane-strided + wave shuffle-tree reduction.
// ---------------------------------------------------------------------------
__global__ void att_scores(const float* __restrict__ enc_proj,
                           const float* __restrict__ dec_proj,
                           const float* __restrict__ att_W,
                           const float* __restrict__ att_b,
                           float* __restrict__ att) {
  const int wid  = (blockIdx.x * blockDim.x + threadIdx.x) >> 5;
  const int lane = threadIdx.x & 31;
  if (wid >= B_ * P_) return;
  const int b = wid / P_;
  const float* __restrict__ ep = enc_proj + (size_t)wid * A_;
  const float* __restrict__ dp = dec_proj + (size_t)b * A_;
  float s = 0.0f;
#pragma unroll 4
  for (int a = lane; a < A_; a += 32)
    s += tanhf(ep[a] + dp[a]) * att_W[a];
#pragma unroll
  for (int off = 16; off > 0; off >>= 1)
    s += __shfl_down(s, off, 32);
  if (lane == 0) att[wid] = s + att_b[0];
}

// ---------------------------------------------------------------------------
// Per-batch softmax over p=196, then context[b,d] = sum_p alpha[p]*enc[b,p,d].
// ---------------------------------------------------------------------------
__global__ void softmax_context(const float* __restrict__ att,
                                const float* __restrict__ enc,
                                float* __restrict__ context) {
  __shared__ float alpha[P_];
  __shared__ float red[256];
  const int b   = blockIdx.x;
  const int tid = threadIdx.x;

  float m = -1e30f;
  for (int p = tid; p < P_; p += blockDim.x)
    m = fmaxf(m, att[b * P_ + p]);
  red[tid] = m;
  __syncthreads();
  for (int s = 128; s > 0; s >>= 1) {
    if (tid < s) red[tid] = fmaxf(red[tid], red[tid + s]);
    __syncthreads();
  }
  m = red[0];
  __syncthreads();

  float sum = 0.0f;
  for (int p = tid; p < P_; p += blockDim.x) {
    float e = __expf(att[b * P_ + p] - m);
    alpha[p] = e;
    sum += e;
  }
  red[tid] = sum;
  __syncthreads();
  for (int s = 128; s > 0; s >>= 1) {
    if (tid < s) red[tid] += red[tid + s];
    __syncthreads();
  }
  const float inv = 1.0f / red[0];
  __syncthreads();
  for (int p = tid; p < P_; p += blockDim.x) alpha[p] *= inv;
  __syncthreads();

  const float* __restrict__ eb = enc + (size_t)b * P_ * ENC_;
  for (int d = tid; d < ENC_; d += blockDim.x) {
    float acc = 0.0f;
#pragma unroll 4
    for (int p = 0; p < P_; ++p)
      acc = fmaf(alpha[p], eb[(size_t)p * ENC_ + d], acc);
    context[b * ENC_ + d] = acc;
  }
}

// ---------------------------------------------------------------------------
__global__ void lstm_pointwise(const float* __restrict__ gates,
                               float* __restrict__ h,
                               float* __restrict__ c) {
  const int idx = blockIdx.x * blockDim.x + threadIdx.x;
  if (idx >= B_ * H_) return;
  const int b = idx / H_;
  const int n = idx % H_;
  const float* __restrict__ g = gates + (size_t)b * 4 * H_;
  const float gi = g[n];
  const float gf = g[H_ + n];
  const float gg = g[2 * H_ + n];
  const float go = g[3 * H_ + n];
  const float c2 = sigmoidf_(gf) * c[idx] + sigmoidf_(gi) * tanhf(gg);
  const float h2 = sigmoidf_(go) * tanhf(c2);
  c[idx] = c2;
  h[idx] = h2;
}

__global__ void zero_hc(float* __restrict__ h, float* __restrict__ c) {
  const int idx = blockIdx.x * blockDim.x + threadIdx.x;
  if (idx < B_ * H_) { h[idx] = 0.0f; c[idx] = 0.0f; }
}

// ---------------------------------------------------------------------------
extern "C" void kernel_launch(void* const* d_in, const int* in_sizes, int n_in,
                              void* d_out, int out_size, void* d_ws, size_t ws_size,
                              hipStream_t stream) {
  (void)in_sizes; (void)n_in; (void)out_size; (void)ws_size;
  const float* encoder_out = (const float*)d_in[0];
  const int*   captions    = (const int*)  d_in[1];
  const float* emb         = (const float*)d_in[2];
  const float* enc_W       = (const float*)d_in[3];
  const float* enc_b       = (const float*)d_in[4];
  const float* dec_W       = (const float*)d_in[5];
  const float* dec_b       = (const float*)d_in[6];
  const float* att_W       = (const float*)d_in[7];
  const float* att_b       = (const float*)d_in[8];
  const float* W_ih        = (const float*)d_in[9];
  const float* W_hh        = (const float*)d_in[10];
  const float* b_ih        = (const float*)d_in[11];
  const float* b_hh        = (const float*)d_in[12];
  const float* fc_W        = (const float*)d_in[13];
  const float* fc_b        = (const float*)d_in[14];
  float* out = (float*)d_out;

  // workspace carve-out (floats): ~13.6 MB total
  float* ws       = (float*)d_ws;
  float* enc_proj = ws; ws += (size_t)B_ * P_ * A_;
  float* h        = ws; ws += B_ * H_;
  float* c        = ws; ws += B_ * H_;
  float* dec_proj = ws; ws += B_ * A_;
  float* att      = ws; ws += B_ * P_;
  float* context  = ws; ws += B_ * ENC_;
  float* gates    = ws; ws += B_ * 4 * H_;

  zero_hc<<<(B_ * H_ + 255) / 256, 256, 0, stream>>>(h, c);

  // one-time: enc_proj = encoder_out @ enc_W.T + enc_b  (6272x512, K=2048)
  // async-LDS staged: 392 strips x 4 N-blocks = 1568 blocks of 8 waves
  {
    const int nblocks = ((B_ * P_) / 16) * ((A_ / 16) / 8);
    gemm_wmma_f32_lds<<<nblocks, 256, 0, stream>>>(
        encoder_out, enc_W, enc_b, enc_proj, B_ * P_, A_, ENC_, A_);
  }

  for (int t = 0; t < TS_; ++t) {
    // dec_proj = h @ dec_W.T + dec_b  (32x512, K=512) -> 64 tiles
    gemm_wmma_f32<<<8, 256, 0, stream>>>(h, dec_W, dec_b, dec_proj,
                                         B_, A_, H_, A_);
    att_scores<<<(B_ * P_ * 32 + 255) / 256, 256, 0, stream>>>(
        enc_proj, dec_proj, att_W, att_b, att);
    softmax_context<<<B_, 256, 0, stream>>>(att, encoder_out, context);
    gates_wmma_f32<<<32, 256, 0, stream>>>(emb, captions, context, h,
                                           W_ih, W_hh, b_ih, b_hh, gates, t);
    lstm_pointwise<<<(B_ * H_ + 255) / 256, 256, 0, stream>>>(gates, h, c);
    // logits = h @ fc_W.T + fc_b  (32x10000, K=512) -> 1250 tiles
    {
      const int tiles = (B_ / 16) * (V_ / 16);
      gemm_wmma_f32<<<(tiles + 7) / 8, 256, 0, stream>>>(
          h, fc_W, fc_b, out + (size_t)t * V_, B_, V_, H_, TS_ * V_);
    }
  }
}